// Patch_6279242187487
// MI455X (gfx1250) — compile-verified
//
#include <hip/hip_runtime.h>
#include <math.h>

typedef __attribute__((ext_vector_type(4))) float        v4f;
typedef __attribute__((ext_vector_type(4))) unsigned int v4u;
typedef __attribute__((ext_vector_type(4))) int          v4i;
typedef __attribute__((ext_vector_type(8))) int          v8i;

#define TILE_BYTES (64 * 1024)   // one full 128x128 fp32 patch

// One workgroup per patch.
// Fast path: TDM (tensor_load_to_lds) DMAs the strided width x width tile
// from the image into LDS (address generation in the DMA engine), then all
// waves stream the contiguous LDS tile to global with NT b128 stores.
// Fallback: direct strided copy (round-1 path).
__global__ __launch_bounds__(256) void patch_gather_tdm(
    const float* __restrict__ img,
    const int*   __restrict__ pos,
    const int*   __restrict__ wptr,
    float*       __restrict__ out,
    int W)
{
    __shared__ float tile[TILE_BYTES / 4];

    const int n     = (int)blockIdx.x;
    const int width = wptr[0];                  // uniform scalar load
    const int py    = pos[2 * n + 0];
    const int px    = pos[2 * n + 1];

    const float* __restrict__ src = img + (size_t)py * (size_t)W + (size_t)px;
    float*       __restrict__ dst = out + (size_t)n  * (size_t)width * (size_t)width;

    const int tid = (int)threadIdx.x;
    const int nt  = (int)blockDim.x;

    if (((width & 3) == 0) && ((width * width * 4) <= TILE_BYTES)) {
        // ---- TDM path -------------------------------------------------
        if ((tid >> 5) == 0) {   // only wave 0 issues the DMA (EXEC ignored by TDM)
            const unsigned long long ga  = (unsigned long long)(uintptr_t)src;
            // Low 32 bits of a flat shared-aperture address == LDS byte offset.
            const unsigned int       lds = (unsigned int)(uintptr_t)(&tile[0]);

            const unsigned int td0 = 1u << 30;  // tensor_dim0: huge -> no OOB zeroing
            const unsigned int td1 = 1u << 30;  // tensor_dim1: huge -> no OOB zeroing
            const unsigned int t0  = (unsigned int)width;  // tile_dim0 (elements/row)
            const unsigned int t1  = (unsigned int)width;  // tile_dim1 (rows)
            const unsigned int str = (unsigned int)W;      // tensor_dim0_stride (elements)

            // D# group 0 (128b): count=1 | lds_addr | global_addr[56:0] | type=2
            v4u g0;
            g0.x = 1u;                                   // count=1, no gather/restore
            g0.y = lds;                                  // lds_addr
            g0.z = (unsigned int)(ga & 0xFFFFFFFFull);   // global_addr[31:0]
            g0.w = (unsigned int)((ga >> 32) & 0x1FFFFFFull) | (2u << 30); // [56:32]|type=2

            // D# group 1 (256b)
            v8i g1;
            g1[0] = (int)(2u << 16);                     // wg_mask=0, data_size=2 (4B), no pad/iterate
            g1[1] = (int)((td0 & 0xFFFFu) << 16);        // atomic_barrier_addr=0 | tensor_dim0[15:0]
            g1[2] = (int)((td0 >> 16) | ((td1 & 0xFFFFu) << 16)); // td0[31:16] | td1[15:0]
            g1[3] = (int)((td1 >> 16) | (t0 << 16));     // td1[31:16] | tile_dim0
            g1[4] = (int)(t1 & 0xFFFFu);                 // tile_dim1 | tile_dim2=0
            g1[5] = (int)str;                            // tensor_dim0_stride[31:0]
            g1[6] = 0;                                   // stride[47:32]=0 | dim1_stride[15:0]=0
            g1[7] = 0;                                   // dim1_stride[47:16]=0

            // Groups 2/3 zero: 2-D tensor (tile_dim2=0 => unused dims).
            v4i gz = {0, 0, 0, 0};

#if defined(__clang_major__) && (__clang_major__ >= 23)
            v8i gz8 = {0, 0, 0, 0, 0, 0, 0, 0};
            __builtin_amdgcn_tensor_load_to_lds(g0, g1, gz, gz, gz8, 0);
#else
            __builtin_amdgcn_tensor_load_to_lds(g0, g1, gz, gz, 0);
#endif
            __builtin_amdgcn_s_wait_tensorcnt(0);
        }
        __syncthreads();   // release all waves once LDS tile is resident

        // Contiguous LDS -> global: ds_load_b128 + NT global_store_b128.
        const int total4 = (width * width) >> 2;
        const v4f* __restrict__ ls = (const v4f*)tile;
        v4f*       __restrict__ gd = (v4f*)dst;
        for (int f = tid; f < total4; f += nt) {
            __builtin_nontemporal_store(ls[f], &gd[f]);
        }
    } else {
        // ---- Fallback: direct strided copy ----------------------------
        int r = tid / width;
        int c = tid - r * width;
        const int dr = nt / width;
        const int dc = nt - dr * width;
        while (r < width) {
            const float val = src[(size_t)r * (size_t)W + c];
            __builtin_nontemporal_store(val, dst + (size_t)r * (size_t)width + c);
            c += dc;
            r += dr;
            if (c >= width) { c -= width; r += 1; }
        }
    }
}

extern "C" void kernel_launch(void* const* d_in, const int* in_sizes, int n_in,
                              void* d_out, int out_size, void* d_ws, size_t ws_size,
                              hipStream_t stream) {
    (void)n_in; (void)out_size; (void)d_ws; (void)ws_size;

    const float* img  = (const float*)d_in[0];   // images: (H, W) float32
    const int*   pos  = (const int*)d_in[1];     // positions: (N, 2) int32
    const int*   wptr = (const int*)d_in[2];     // width: scalar int32 on device
    float*       out  = (float*)d_out;           // (N, width, width) float32

    const int N = in_sizes[1] / 2;
    const int W = (int)(sqrt((double)in_sizes[0]) + 0.5);   // square image

    patch_gather_tdm<<<N, 256, 0, stream>>>(img, pos, wptr, out, W);
}